// LbpBlock_31636729102830
// MI455X (gfx1250) — compile-verified
//
#include <hip/hip_runtime.h>

typedef float v2f __attribute__((ext_vector_type(2)));
typedef float v8f __attribute__((ext_vector_type(8)));

#define TILE 16
#define HALO 18
#define NPIX (HALO * HALO)   // 324 halo pixels
#define NGROUP 21            // ceil(324/16)
#define ROWPAD 337           // LDS row stride (words), breaks bank conflicts
#define OC 64
#define OC_HALF 32

__global__ __launch_bounds__(256) void lbp_fused_kernel(
    const float* __restrict__ x,       // (8,3,256,256)
    const float* __restrict__ conv_w,  // (64,3)
    const float* __restrict__ w,       // (64,8)
    float* __restrict__ out)           // (8,67,256,256)
{
  __shared__ float y_lds[OC_HALF][ROWPAD];        // ~43 KB
  __shared__ __align__(16) float ew[OC][8];       // 2 KB

  const int n    = blockIdx.z;
  const int th   = blockIdx.y * TILE;
  const int tw   = blockIdx.x * TILE;
  const int tid  = threadIdx.x;     // 0..255
  const int lane = tid & 31;
  // force scalar wave id so the WMMA unit loop is uniform (EXEC stays all-ones)
  const int wv   = __builtin_amdgcn_readfirstlane(tid >> 5);
  const int pl   = lane & 15;
  const bool lo  = lane < 16;
  const size_t img = 256 * 256;

  // exp(w) into LDS (512 values)
  for (int i = tid; i < OC * 8; i += 256)
    ew[i >> 3][i & 7] = __expf(w[i]);

  // this thread's output pixel
  const int ty   = tid >> 4;        // 0..15
  const int tx   = tid & 15;
  const int h    = th + ty;
  const int wc   = tw + tx;
  const bool border = (h == 0) | (h == 255) | (wc == 0) | (wc == 255);

  // passthrough channels 0..2
  {
    const size_t pix = (size_t)h * 256 + wc;
    #pragma unroll
    for (int c = 0; c < 3; ++c)
      out[((size_t)n * 67 + c) * img + pix] = x[((size_t)n * 3 + c) * img + pix];
  }

  const int h0 = th - 1, w0 = tw - 1;  // halo origin
  const size_t n3img = (size_t)n * 3 * img;

  for (int half = 0; half < 2; ++half) {
    const int obase = half * OC_HALF;

    // A operands for both 16-channel o-blocks of this half (branchless, hoisted)
    // A layout 16x4 f32: v0 = K0 (lanes 0-15) / K2 (lanes 16-31); v1 = K1 / K3(=0)
    v2f a0, a1;
    {
      const int m0 = (obase + pl) * 3;
      const int m1 = (obase + 16 + pl) * 3;
      const float c00 = conv_w[m0], c01 = conv_w[m0 + 1], c02 = conv_w[m0 + 2];
      const float c10 = conv_w[m1], c11 = conv_w[m1 + 1], c12 = conv_w[m1 + 2];
      a0.x = lo ? c00 : c02;  a0.y = lo ? c01 : 0.0f;
      a1.x = lo ? c10 : c12;  a1.y = lo ? c11 : 0.0f;
    }

    // ---- y halo tile via V_WMMA_F32_16X16X4_F32 (scalar-uniform unit loop) ----
    for (int unit = wv; unit < NGROUP * 2; unit += 8) {
      const int g  = unit >> 1;
      const int ob = unit & 1;                    // 0 or 1 -> o-block

      // B: 4x16 f32 (K=c, N=pixel), clamped halo coords; all loads unconditional
      int p = g * 16 + pl; p = (p > NPIX - 1) ? (NPIX - 1) : p;
      const int r  = p / HALO, c = p % HALO;
      int gh = h0 + r; gh = gh < 0 ? 0 : (gh > 255 ? 255 : gh);
      int gw = w0 + c; gw = gw < 0 ? 0 : (gw > 255 ? 255 : gw);
      const size_t base = n3img + (size_t)gh * 256 + gw;
      const float xc0 = x[base + (lo ? (size_t)0 : 2 * img)];  // c0 or c2
      const float xc1 = x[base + img];                         // c1 (always valid)
      v2f b;
      b.x = xc0;
      b.y = lo ? xc1 : 0.0f;

      const v2f a = ob ? a1 : a0;
      v8f d = {};
      d = __builtin_amdgcn_wmma_f32_16x16x4_f32(
              /*neg_a=*/false, a, /*neg_b=*/false, b,
              /*c_mod=*/(short)0, d, /*reuse_a=*/false, /*reuse_b=*/false);

      // D layout: lanes 0-15 hold rows oblk..+7, lanes 16-31 rows oblk+8..+15
      const int orow = ob * 16 + (lo ? 0 : 8);
      const int pp   = g * 16 + pl;   // may reach 335 < ROWPAD (padding, never read)
      #pragma unroll
      for (int rr = 0; rr < 8; ++rr)
        y_lds[orow + rr][pp] = d[rr];
    }
    __syncthreads();

    // ---- LBP threshold + weighted sum out of LDS ----
    const size_t obaddr = ((size_t)n * 67 + 3 + obase) * img + (size_t)h * 256 + wc;
    if (!border) {
      const int pc = (ty + 1) * HALO + (tx + 1);
      // reference shift order: UL, U, UR, L, DL, D, DR, R
      const int off[8] = { -HALO - 1, -HALO, -HALO + 1, -1,
                            HALO - 1,  HALO,  HALO + 1, +1 };
      for (int o = 0; o < OC_HALF; ++o) {
        const float yc = y_lds[o][pc];
        const float4 e0 = *(const float4*)&ew[obase + o][0];   // ds_load_b128
        const float4 e1 = *(const float4*)&ew[obase + o][4];   // ds_load_b128
        const float ewv[8] = { e0.x, e0.y, e0.z, e0.w, e1.x, e1.y, e1.z, e1.w };
        float acc = 0.0f;
        #pragma unroll
        for (int l = 0; l < 8; ++l) {
          const float s = y_lds[o][pc + off[l]];
          acc += (yc - s > 0.0f) ? ewv[l] : 0.0f;
        }
        out[obaddr + (size_t)o * img] = acc;
      }
    } else {
      for (int o = 0; o < OC_HALF; ++o)
        out[obaddr + (size_t)o * img] = 0.0f;
    }
    __syncthreads();
  }
}

extern "C" void kernel_launch(void* const* d_in, const int* in_sizes, int n_in,
                              void* d_out, int out_size, void* d_ws, size_t ws_size,
                              hipStream_t stream) {
  const float* x      = (const float*)d_in[0];  // (8,3,256,256)
  const float* conv_w = (const float*)d_in[1];  // (64,3)
  const float* w      = (const float*)d_in[2];  // (64,8)
  float* out          = (float*)d_out;          // (8,67,256,256)
  (void)in_sizes; (void)n_in; (void)out_size; (void)d_ws; (void)ws_size;

  dim3 grid(256 / TILE, 256 / TILE, 8);  // 16 x 16 x 8 workgroups
  dim3 block(256);
  lbp_fused_kernel<<<grid, block, 0, stream>>>(x, conv_w, w, out);
}